// ConvSkip_87488483819569
// MI455X (gfx1250) — compile-verified
//
#include <hip/hip_runtime.h>

typedef __attribute__((ext_vector_type(2))) float v2f;
typedef __attribute__((ext_vector_type(4))) float v4f;
typedef __attribute__((ext_vector_type(8))) float v8f;

#define N_NODES    50000
#define DEG        16
#define IN_SIZE    128
#define OUT_SIZE   64
#define MERGE_SIZE 128
#define NTILES     (N_NODES / 16)      // 3125 (exact)
#define WPB        8                   // waves per block
#define BLOCK      256

// Swizzled-B LDS layout: for K-step kq (0..31), half 'hi' (K rows kq*4+2*hi, +1),
// column c (0..15), the 8 floats a lane consumes are contiguous:
//   entry(kq,hi,c) = [ W[k][c], W[k+1][c],  W[k][c+16], W[k+1][c+16],
//                      W[k][c+32], W[k+1][c+32], W[k][c+48], W[k+1][c+48] ]
// Entry stride padded 8 -> 12 floats (48 B): bank step 12 (conflict-free mod 64),
// still 16-byte aligned for ds_load_b128.
#define ENTRY_F    12
#define ROW_F      (16 * ENTRY_F)              // 192 floats per (kq,hi) row
#define SWZ_FLOATS (64 * ROW_F)                // 32 kq * 2 hi * 192 = 12288 (48 KB)

__device__ __forceinline__ void fill_swizzled_W(const float* __restrict__ W, float* sW) {
    for (int q = threadIdx.x; q < (IN_SIZE * OUT_SIZE) / 4; q += BLOCK) {
        v4f w = ((const v4f*)W)[q];                    // coalesced b128 read
        int base = q * 4;
        #pragma unroll
        for (int e = 0; e < 4; ++e) {
            int idx = base + e;
            int k   = idx >> 6;                        // K row (0..127)
            int j   = idx & 63;                        // N col (0..63)
            int kq  = k >> 2;
            int rem = k & 3;
            int hi2 = rem >> 1;
            int rr  = rem & 1;
            int n   = j >> 4;
            int c   = j & 15;
            sW[(kq * 2 + hi2) * ROW_F + c * ENTRY_F + n * 2 + rr] = w[e];
        }
    }
}

// One wave computes a 16x64 fp32 tile: D = A(16x128) * B(128x64) via 32 K-steps
// of V_WMMA_F32_16X16X4_F32. Per K-step: 1 global b64 (A) + 2 ds b128 (B) + 4 WMMA.
__device__ __forceinline__ void wave_gemm_16x64(const float* __restrict__ arow,
                                                const float* sW,
                                                int hi, int col,
                                                v8f& a0, v8f& a1, v8f& a2, v8f& a3) {
    const int khalf = hi * 2;
    #pragma unroll 4
    for (int kq = 0; kq < 32; ++kq) {
        v2f a = *(const v2f*)(arow + kq * 4 + khalf);  // global_load_b64

        const float* bp = sW + (kq * 2 + hi) * ROW_F + col * ENTRY_F;
        v4f q0 = *(const v4f*)(bp);                    // ds_load_b128: b0,b1 pairs
        v4f q1 = *(const v4f*)(bp + 4);                // ds_load_b128: b2,b3 pairs
        v2f b0 = {q0.x, q0.y}, b1 = {q0.z, q0.w};
        v2f b2 = {q1.x, q1.y}, b3 = {q1.z, q1.w};

        a0 = __builtin_amdgcn_wmma_f32_16x16x4_f32(false, a, false, b0, (short)0, a0, false, false);
        a1 = __builtin_amdgcn_wmma_f32_16x16x4_f32(false, a, false, b1, (short)0, a1, false, false);
        a2 = __builtin_amdgcn_wmma_f32_16x16x4_f32(false, a, false, b2, (short)0, a2, false, false);
        a3 = __builtin_amdgcn_wmma_f32_16x16x4_f32(false, a, false, b3, (short)0, a3, false, false);
    }
}

// ---------------------------------------------------------------------------
// Kernel 1: h = data @ W_lin + b_lin
// ---------------------------------------------------------------------------
__global__ __launch_bounds__(BLOCK)
void convskip_h_gemm(const float* __restrict__ data,
                     const float* __restrict__ W,      // [128 x 64]
                     const float* __restrict__ bias,   // [64]
                     float* __restrict__ h)            // [N x 64]
{
    __shared__ float sW[SWZ_FLOATS];                   // 48 KB, swizzled
    __shared__ float sB[OUT_SIZE];

    fill_swizzled_W(W, sW);
    if (threadIdx.x < OUT_SIZE) sB[threadIdx.x] = bias[threadIdx.x];
    __syncthreads();

    const int wave = threadIdx.x >> 5;
    const int lane = threadIdx.x & 31;
    const int tile = blockIdx.x * WPB + wave;
    if (tile >= NTILES) return;                        // wave-uniform: EXEC stays full

    const int col = lane & 15;
    const int hi  = lane >> 4;

    const float* arow = data + (size_t)(tile * 16 + col) * IN_SIZE;

    v8f acc0 = {}, acc1 = {}, acc2 = {}, acc3 = {};
    wave_gemm_16x64(arow, sW, hi, col, acc0, acc1, acc2, acc3);

    const float bb0 = sB[col], bb1 = sB[16 + col], bb2 = sB[32 + col], bb3 = sB[48 + col];
    float* hb = h + (size_t)(tile * 16 + hi * 8) * OUT_SIZE + col;
    #pragma unroll
    for (int r = 0; r < 8; ++r) {
        size_t ro = (size_t)r * OUT_SIZE;
        hb[ro +  0] = acc0[r] + bb0;
        hb[ro + 16] = acc1[r] + bb1;
        hb[ro + 32] = acc2[r] + bb2;
        hb[ro + 48] = acc3[r] + bb3;
    }
}

// ---------------------------------------------------------------------------
// Kernel 2: out = relu( (h - neigh_sum/deg) + merge @ W_tr + b_tr )
// ---------------------------------------------------------------------------
__global__ __launch_bounds__(BLOCK)
void convskip_out(const float* __restrict__ merge,
                  const int*   __restrict__ structure, // [N x 16]
                  const float* __restrict__ W,         // [128 x 64]
                  const float* __restrict__ bias,      // [64]
                  const float* __restrict__ h,         // [N x 64]
                  float* __restrict__ out)             // [N x 64]
{
    __shared__ float sW[SWZ_FLOATS];
    __shared__ float sB[OUT_SIZE];

    fill_swizzled_W(W, sW);
    if (threadIdx.x < OUT_SIZE) sB[threadIdx.x] = bias[threadIdx.x];
    __syncthreads();

    const int wave = threadIdx.x >> 5;
    const int lane = threadIdx.x & 31;
    const int tile = blockIdx.x * WPB + wave;
    if (tile >= NTILES) return;

    const int col = lane & 15;
    const int hi  = lane >> 4;

    const float* arow = merge + (size_t)(tile * 16 + col) * MERGE_SIZE;

    v8f acc0 = {}, acc1 = {}, acc2 = {}, acc3 = {};
    wave_gemm_16x64(arow, sW, hi, col, acc0, acc1, acc2, acc3);

    const float bb0 = sB[col], bb1 = sB[16 + col], bb2 = sB[32 + col], bb3 = sB[48 + col];
    const float inv_deg = 1.0f / (float)DEG;

    #pragma unroll
    for (int r = 0; r < 8; ++r) {
        const int node = tile * 16 + hi * 8 + r;
        const int* st  = structure + (size_t)node * DEG;

        float s0 = 0.f, s1 = 0.f, s2 = 0.f, s3 = 0.f;
        #pragma unroll
        for (int j = 0; j < DEG; ++j) {
            unsigned nbr = (unsigned)st[j];
            if (nbr >= (unsigned)N_NODES) nbr = 0;     // guard garbage indices
            const float* hr = h + (size_t)nbr * OUT_SIZE + col;
            s0 += hr[0];  s1 += hr[16];  s2 += hr[32];  s3 += hr[48];
        }

        const float* hn = h + (size_t)node * OUT_SIZE + col;
        float o0 = hn[0]  - s0 * inv_deg + acc0[r] + bb0;
        float o1 = hn[16] - s1 * inv_deg + acc1[r] + bb1;
        float o2 = hn[32] - s2 * inv_deg + acc2[r] + bb2;
        float o3 = hn[48] - s3 * inv_deg + acc3[r] + bb3;

        float* ob = out + (size_t)node * OUT_SIZE + col;
        ob[0]  = o0 > 0.f ? o0 : 0.f;
        ob[16] = o1 > 0.f ? o1 : 0.f;
        ob[32] = o2 > 0.f ? o2 : 0.f;
        ob[48] = o3 > 0.f ? o3 : 0.f;
    }
}

extern "C" void kernel_launch(void* const* d_in, const int* in_sizes, int n_in,
                              void* d_out, int out_size, void* d_ws, size_t ws_size,
                              hipStream_t stream) {
    const float* data      = (const float*)d_in[0];
    const float* merge     = (const float*)d_in[1];
    const int*   structure = (const int*)  d_in[2];
    const float* W_lin     = (const float*)d_in[3];
    const float* b_lin     = (const float*)d_in[4];
    const float* W_tr      = (const float*)d_in[5];
    const float* b_tr      = (const float*)d_in[6];
    float* out = (float*)d_out;
    float* h   = (float*)d_ws;                        // 50000*64*4 = 12.8 MB

    const int grid = (NTILES + WPB - 1) / WPB;        // 391 blocks of 8 waves
    convskip_h_gemm<<<grid, BLOCK, 0, stream>>>(data, W_lin, b_lin, h);
    convskip_out  <<<grid, BLOCK, 0, stream>>>(merge, structure, W_tr, b_tr, h, out);
}